// SpodNet_61692910239924
// MI455X (gfx1250) — compile-verified
//
#include <hip/hip_runtime.h>
#include <cmath>

#define P     128
#define PT    129          // padded LDS row stride (odd -> bank-conflict-free column access)
#define NTHR  256          // 8 wave32 waves
#define NSTEP 256          // K * P
#define DIAG  0.1f

typedef __attribute__((ext_vector_type(2))) float v2f;
typedef __attribute__((ext_vector_type(8))) float v8f;

// y[0..127] = op(A(mdim x kdim, row stride lda) * x + biasLds), via V_WMMA_F32_16X16X4_F32.
// Wave w computes rows [16w, 16w+16); vector lives in column 0 of the B fragment.
// All loads are unconditional (clamped indices + VALU selects) so the scheduler can
// batch ds/global loads ahead of the WMMA chain instead of exec-predicating them.
// Must be called by all 256 threads (EXEC all-ones around the WMMA).
__device__ __forceinline__ void wmma_matvec128(
    const float* __restrict__ A, int lda, int kdim,
    const float* __restrict__ xv, float* __restrict__ yv,
    const float* __restrict__ biasLds, bool relu, int tid)
{
  const int wave = tid >> 5;
  const int lane = tid & 31;
  const int m    = lane & 15;
  const int hi   = lane >> 4;          // 0: K pair {0,1}; 1: K pair {2,3}
  const int row  = wave * 16 + m;      // callers guarantee A has >= 128 addressable rows
  const int kc   = kdim - 1;
  v8f acc = {0.f, 0.f, 0.f, 0.f, 0.f, 0.f, 0.f, 0.f};
  #pragma unroll
  for (int k0 = 0; k0 < P; k0 += 4) {
    const int ka = k0 + 2 * hi;
    // A 16x4 fragment: lanes 0-15 hold K={0,1}, lanes 16-31 hold K={2,3}
    const int i0 = ka     <= kc ? ka     : kc;   // clamp -> unconditional in-bounds load
    const int i1 = ka + 1 <= kc ? ka + 1 : kc;
    float a0 = A[row * lda + i0];
    float a1 = A[row * lda + i1];
    // B 4x16 fragment, only column N=0 carries the vector (xv is zero-padded)
    const float t0 = xv[k0 + hi];        // 2-address LDS broadcast, no predication
    const float t1 = xv[k0 + 2 + hi];
    v2f a, b;
    a.x = (ka     <= kc) ? a0 : 0.0f;    // v_cndmask, not exec-masked load
    a.y = (ka + 1 <= kc) ? a1 : 0.0f;
    b.x = (m == 0) ? t0 : 0.0f;
    b.y = (m == 0) ? t1 : 0.0f;
    acc = __builtin_amdgcn_wmma_f32_16x16x4_f32(
        false, a, false, b, (short)0, acc, false, false);
  }
  // D column 0: lane 0 has M=0..7 in acc[0..7], lane 16 has M=8..15
  if (m == 0) {
    const int base = wave * 16 + hi * 8;
    #pragma unroll
    for (int j = 0; j < 8; ++j) {
      const int rr = base + j;
      float v = acc[j];
      if (biasLds) v += biasLds[rr];     // zero-padded LDS bias, unconditional
      if (relu) v = fmaxf(v, 0.0f);
      yv[rr] = v;
    }
  }
}

__global__ void __launch_bounds__(NTHR)
spodnet_kernel(const float* __restrict__ S,
               const float* __restrict__ a_w1, const float* __restrict__ a_b1,
               const float* __restrict__ a_w2, const float* __restrict__ a_b2,
               const float* __restrict__ a_w3, const float* __restrict__ a_b3,
               const float* __restrict__ c_w1, const float* __restrict__ c_b1,
               const float* __restrict__ c_w2, const float* __restrict__ c_b2,
               const float* __restrict__ c_w3, const float* __restrict__ c_b3,
               float* __restrict__ Wout)
{
  extern __shared__ float smem[];
  float* Tsh = smem;               // P*PT : Theta (padded)
  float* Wsh = Tsh + P * PT;       // P*PT : W     (padded)
  float* vwc = Wsh + P * PT;       // compact w12 (127 + zero pad)
  float* vh1 = vwc + P;
  float* vh2 = vh1 + P;
  float* vwn = vh2 + P;            // compact w12_next
  float* vx  = vwn + P;            // full-space embedding of w12_next
  float* vy  = vx  + P;            // Theta * x
  float* va  = vy  + P;            // theta_12 (full space, a[c]=0)
  float* vu  = va  + P;            // inv_W11 * w (full space, u[c]=0)
  float* b1s = vu  + P;            // LDS copies of biases (zero-padded)
  float* b2s = b1s + P;
  float* b3s = b2s + P;
  float* sc  = b3s + P;            // broadcast scalars [8]

  const int tid = threadIdx.x;
  const int b   = blockIdx.x;
  const float* Sg = S    + (size_t)b * P * P;
  float*       Wg = Wout + (size_t)b * P * P;

  // ---- init: W0 = S + 0.1 I into both Theta-work and W (LDS); biases to LDS ----
  for (int e = tid; e < P * P; e += NTHR) {
    const int r = e >> 7, cc = e & (P - 1);
    const float v = Sg[e] + (r == cc ? DIAG : 0.0f);
    Tsh[r * PT + cc] = v;
    Wsh[r * PT + cc] = v;
  }
  if (tid < P) {
    b1s[tid] = c_b1[tid];
    b2s[tid] = c_b2[tid];
    b3s[tid] = (tid < P - 1) ? c_b3[tid] : 0.0f;
  }
  __syncthreads();

  // ---- Gauss-Jordan in-place inversion: Tsh <- W0^{-1} (SPD, no pivoting) ----
  const int r2 = tid >> 1;               // each row handled by 2 threads
  const int j0 = (tid & 1) * (P / 2);    // column half
  for (int k = 0; k < P; ++k) {
    const float invp = 1.0f / Tsh[k * PT + k];
    const float f    = Tsh[r2 * PT + k];
    __syncthreads();
    if (tid < P) {
      if (tid == k) Tsh[k * PT + k] = invp;
      else          Tsh[k * PT + tid] *= invp;
    }
    __syncthreads();
    if (r2 != k) {
      for (int j = j0; j < j0 + P / 2; ++j)
        if (j != k) Tsh[r2 * PT + j] -= f * Tsh[k * PT + j];
      if (k >= j0 && k < j0 + P / 2) Tsh[r2 * PT + k] = -f * invp;
    }
    __syncthreads();
  }

  // ---- 256 sequential column-update steps ----
  for (int it = 0; it < NSTEP; ++it) {
    const int c = it & (P - 1);

    // compact w12[t] = W[idx[t], c]
    if (tid < P) {
      if (tid < P - 1) {
        const int j = tid + (tid >= c ? 1 : 0);
        vwc[tid] = Wsh[j * PT + c];
      } else vwc[P - 1] = 0.0f;
    }
    __syncthreads();

    // col_mlp: three matvecs on the matrix pipe (weights stream from L2)
    wmma_matvec128(c_w1, P - 1, P - 1, vwc, vh1, b1s, true,  tid);
    __syncthreads();
    wmma_matvec128(c_w2, P,     P,     vh1, vh2, b2s, true,  tid);
    __syncthreads();
    wmma_matvec128(c_w3, P,     P,     vh2, vwn, b3s, false, tid);  // row 127 garbage, unused
    __syncthreads();

    // scatter w12_next into full space; capture theta_12 before Theta changes
    if (tid < P) {
      if (tid < P - 1) {
        const int j = tid + (tid >= c ? 1 : 0);
        vx[j] = vwn[tid];
      } else vx[c] = 0.0f;
      va[tid] = (tid == c) ? 0.0f : Tsh[tid * PT + c];
    }
    __syncthreads();

    // y = Theta * x  (LDS-resident Theta through WMMA)
    wmma_matvec128(Tsh, PT, P, vx, vy, nullptr, false, tid);
    __syncthreads();

    // reductions + scalar chain (wave 0)
    if (tid < 32) {
      float s = 0.0f;
      for (int q = tid; q < P; q += 32) s += vx[q] * vy[q];
      #pragma unroll
      for (int off = 16; off > 0; off >>= 1) s += __shfl_down(s, off, 32);
      if (tid == 0) {
        const float t22  = Tsh[c * PT + c];
        const float s1   = vy[c];                  // theta_12 . w
        const float zeta = s - s1 * s1 / t22;      // w^T inv_W11 w
        // alpha_mlp(w22): 1 -> 3 -> 3 -> 1, exp output
        const float w22 = Wsh[c * PT + c];
        float h[3], g[3];
        #pragma unroll
        for (int i = 0; i < 3; ++i) h[i] = fmaxf(a_w1[i] * w22 + a_b1[i], 0.0f);
        #pragma unroll
        for (int i = 0; i < 3; ++i) {
          float t = a_b2[i];
          #pragma unroll
          for (int j = 0; j < 3; ++j) t += a_w2[i * 3 + j] * h[j];
          g[i] = fmaxf(t, 0.0f);
        }
        float al = a_b3[0];
        #pragma unroll
        for (int j = 0; j < 3; ++j) al += a_w3[j] * g[j];
        const float alpha = __expf(al);
        sc[0] = s1 / t22;          // u coefficient
        sc[1] = -1.0f / t22;       // rank-2 coef on a a^T
        sc[2] = 1.0f / alpha;      // theta_22_next == rank-2 coef on u u^T
        sc[3] = alpha + zeta;      // w_22_next
      }
    }
    __syncthreads();

    if (tid < P)
      vu[tid] = (tid == c) ? 0.0f : (vy[tid] - sc[0] * va[tid]);
    __syncthreads();

    // rank-2 Theta update: T += (-1/t22) a a^T + t22n u u^T  (row/col c no-op: a[c]=u[c]=0)
    {
      const float ca = sc[1] * va[r2];
      const float cu = sc[2] * vu[r2];
      for (int j = j0; j < j0 + P / 2; ++j)
        Tsh[r2 * PT + j] += ca * va[j] + cu * vu[j];
    }
    __syncthreads();

    // explicit Theta row/col c and W row/col/diag c
    if (tid < P) {
      if (tid == c) {
        Tsh[c * PT + c] = sc[2];
        Wsh[c * PT + c] = sc[3];
      } else {
        const float t12n = -sc[2] * vu[tid];
        Tsh[tid * PT + c] = t12n;
        Tsh[c * PT + tid] = t12n;
      }
      if (tid < P - 1) {
        const int j = tid + (tid >= c ? 1 : 0);
        const float v = vwn[tid];
        Wsh[j * PT + c] = v;
        Wsh[c * PT + j] = v;
      }
    }
    __syncthreads();
  }

  // ---- write final W ----
  for (int e = tid; e < P * P; e += NTHR)
    Wg[e] = Wsh[(e >> 7) * PT + (e & (P - 1))];
}

extern "C" void kernel_launch(void* const* d_in, const int* in_sizes, int n_in,
                              void* d_out, int out_size, void* d_ws, size_t ws_size,
                              hipStream_t stream) {
  (void)n_in; (void)out_size; (void)d_ws; (void)ws_size;
  const float* S    = (const float*)d_in[0];
  const float* a_w1 = (const float*)d_in[1];
  const float* a_b1 = (const float*)d_in[2];
  const float* a_w2 = (const float*)d_in[3];
  const float* a_b2 = (const float*)d_in[4];
  const float* a_w3 = (const float*)d_in[5];
  const float* a_b3 = (const float*)d_in[6];
  const float* c_w1 = (const float*)d_in[7];
  const float* c_b1 = (const float*)d_in[8];
  const float* c_w2 = (const float*)d_in[9];
  const float* c_b2 = (const float*)d_in[10];
  const float* c_w3 = (const float*)d_in[11];
  const float* c_b3 = (const float*)d_in[12];
  float* Wout = (float*)d_out;

  const int Bn = in_sizes[0] / (P * P);   // 512
  const size_t smem = (size_t)(2 * P * PT + 11 * P + 8) * sizeof(float); // ~138 KB

  // capture-safe host-side attribute set (idempotent, deterministic)
  (void)hipFuncSetAttribute((const void*)spodnet_kernel,
                            hipFuncAttributeMaxDynamicSharedMemorySize, (int)smem);

  spodnet_kernel<<<dim3(Bn), dim3(NTHR), smem, stream>>>(
      S, a_w1, a_b1, a_w2, a_b2, a_w3, a_b3,
      c_w1, c_b1, c_w2, c_b2, c_w3, c_b3, Wout);
}